// ConvAttentionLayer_33225867002152
// MI455X (gfx1250) — compile-verified
//
#include <hip/hip_runtime.h>
#include <hip/hip_bf16.h>
#include <math.h>

typedef __attribute__((ext_vector_type(2))) float v2f;
typedef __attribute__((ext_vector_type(8))) float v8f;

#define IN_DIM 128
#define OUT_DIM 64
#define BN_EPS 1e-5f

// ---------------------------------------------------------------------------
// K0: init workspace (segment max = -inf, denom = 0, agg = 0)
// ---------------------------------------------------------------------------
__global__ void init_ws_kernel(float* __restrict__ m, float* __restrict__ denom,
                               float* __restrict__ agg, int N) {
    int idx = blockIdx.x * blockDim.x + threadIdx.x;
    int total = N * OUT_DIM;
    if (idx < total) agg[idx] = 0.0f;
    if (idx < N) {
        m[idx] = -INFINITY;
        denom[idx] = 0.0f;
    }
}

// ---------------------------------------------------------------------------
// K1: inp = input(N x 128) @ W(128 x 64), f32 WMMA 16x16x4.
// Block = 256 threads = 8 waves; each wave owns a 16-row x 64-col strip
// (4 accumulator tiles), K-loop 128/4 = 32 WMMA steps per tile.
//
// A frag (16x4, ISA 7.12.2): lanes 0-15 hold K={kk,kk+1} for M=lane;
// lanes 16-31 hold K={kk+2,kk+3}. k0 is even -> one global_load_b64.
//
// B frag (4x16): same K split, N = (lane&15) + tile offset. W is staged in
// LDS in K-PAIRED layout sB2[k0/2][col] = {W[k0][col], W[k0+1][col]} so each
// fragment is a single aligned ds_load_b64 straight into the even VGPR pair
// (no mov repacking before v_wmma).
//
// C/D (16x16 f32): vgpr v -> row v (lanes 0-15) / row v+8 (lanes 16-31).
// ---------------------------------------------------------------------------
__global__ void gemm_wmma_kernel(const float* __restrict__ A,
                                 const float* __restrict__ B,
                                 float* __restrict__ C, int M) {
    __shared__ v2f sB2[(IN_DIM / 2) * OUT_DIM];   // 32 KB of 320 KB WGP LDS
    for (int p = threadIdx.x; p < (IN_DIM / 2) * OUT_DIM; p += blockDim.x) {
        const int k   = (p >> 6) << 1;   // even K row
        const int col = p & 63;
        v2f w;
        w.x = B[k * OUT_DIM + col];
        w.y = B[(k + 1) * OUT_DIM + col];
        sB2[p] = w;
    }
    __syncthreads();

    const int wave  = threadIdx.x >> 5;
    const int lane  = threadIdx.x & 31;
    const int l16   = lane & 15;      // row within A tile / col within B tile
    const int khalf = lane >> 4;      // which K pair this lane holds

    const int rowBase = (blockIdx.x * 8 + wave) * 16;
    if (rowBase >= M) return;

    v8f acc0 = {}, acc1 = {}, acc2 = {}, acc3 = {};
    const float* arow = A + (size_t)(rowBase + l16) * IN_DIM;

    for (int kk = 0; kk < IN_DIM; kk += 4) {
        const int k0 = kk + 2 * khalf;               // even
        const v2f a = *(const v2f*)(arow + k0);      // global_load_b64
        const int pbase = (k0 >> 1) * OUT_DIM + l16; // paired-row base
        const v2f b0 = sB2[pbase +  0];              // ds_load_b64 each
        const v2f b1 = sB2[pbase + 16];
        const v2f b2 = sB2[pbase + 32];
        const v2f b3 = sB2[pbase + 48];
        acc0 = __builtin_amdgcn_wmma_f32_16x16x4_f32(false, a, false, b0, (short)0, acc0, false, false);
        acc1 = __builtin_amdgcn_wmma_f32_16x16x4_f32(false, a, false, b1, (short)0, acc1, false, false);
        acc2 = __builtin_amdgcn_wmma_f32_16x16x4_f32(false, a, false, b2, (short)0, acc2, false, false);
        acc3 = __builtin_amdgcn_wmma_f32_16x16x4_f32(false, a, false, b3, (short)0, acc3, false, false);
    }

#pragma unroll
    for (int v = 0; v < 8; ++v) {
        const size_t r = (size_t)(rowBase + v + 8 * khalf) * OUT_DIM;
        C[r +  0 + l16] = acc0[v];
        C[r + 16 + l16] = acc1[v];
        C[r + 32 + l16] = acc2[v];
        C[r + 48 + l16] = acc3[v];
    }
}

// ---------------------------------------------------------------------------
// float atomic max: signed-int max for non-negative, unsigned min for
// negative. Correct under mixed-sign races when initialized to -inf.
// ---------------------------------------------------------------------------
__device__ __forceinline__ void atomicMaxFloat(float* addr, float val) {
    if (val >= 0.0f)
        atomicMax((int*)addr, __float_as_int(val));
    else
        atomicMin((unsigned int*)addr, __float_as_uint(val));
}

// ---------------------------------------------------------------------------
// K2: per-edge conv+BN+relu+fc score (BN folded into conv coeffs),
// then leaky-relu and segment-max via atomics. One lane per edge.
// Conv layout: x[j,0]=h[j], x[j,1]=r[j], x[j,2]=t[j]; y[c,i] uses window
// j=i..i+2 -> 3-tap sliding registers per stream.
// ---------------------------------------------------------------------------
__global__ void edge_score_kernel(const int* __restrict__ triple,
                                  const float* __restrict__ inp,
                                  const float* __restrict__ rel_embed,
                                  const float* __restrict__ conv_w,
                                  const float* __restrict__ conv_b,
                                  const float* __restrict__ fc_w,
                                  const float* __restrict__ bn1_gamma,
                                  const float* __restrict__ bn1_beta,
                                  const float* __restrict__ bn2_gamma,
                                  const float* __restrict__ bn2_beta,
                                  float* __restrict__ eArr,
                                  float* __restrict__ m, int E) {
    __shared__ float s_coef[8][3][3];
    __shared__ float s_bias[8];
    __shared__ float s_fc[8 * 62];

    const int tid = threadIdx.x;
    if (tid < 8) {
        const float scale1 = bn1_gamma[0] * rsqrtf(1.0f + BN_EPS);
        const float beta1  = bn1_beta[0];
        const float s2     = bn2_gamma[tid] * rsqrtf(1.0f + BN_EPS);
        float sumw = 0.0f;
#pragma unroll
        for (int ky = 0; ky < 3; ++ky)
#pragma unroll
            for (int kx = 0; kx < 3; ++kx) {
                const float w = conv_w[tid * 9 + ky * 3 + kx];
                sumw += w;
                s_coef[tid][ky][kx] = s2 * scale1 * w;
            }
        s_bias[tid] = s2 * (beta1 * sumw + conv_b[tid]) + bn2_beta[tid];
    }
    for (int i = tid; i < 8 * 62; i += blockDim.x) s_fc[i] = fc_w[i];
    __syncthreads();

    const int e = blockIdx.x * blockDim.x + tid;
    if (e >= E) return;

    const int head = triple[3 * e + 0];
    const int rel  = triple[3 * e + 1];
    const int tail = triple[3 * e + 2];

    const float* hp = inp + (size_t)head * OUT_DIM;
    const float* rp = rel_embed + (size_t)rel * OUT_DIM;
    const float* tp = inp + (size_t)tail * OUT_DIM;

    float h0 = hp[0], h1 = hp[1];
    float r0 = rp[0], r1 = rp[1];
    float t0 = tp[0], t1 = tp[1];
    float acc = 0.0f;

    for (int i = 0; i < 62; ++i) {
        const float h2 = hp[i + 2];
        const float r2 = rp[i + 2];
        const float t2 = tp[i + 2];
#pragma unroll
        for (int c = 0; c < 8; ++c) {
            float z = s_bias[c];
            z = fmaf(s_coef[c][0][0], h0, z);
            z = fmaf(s_coef[c][0][1], r0, z);
            z = fmaf(s_coef[c][0][2], t0, z);
            z = fmaf(s_coef[c][1][0], h1, z);
            z = fmaf(s_coef[c][1][1], r1, z);
            z = fmaf(s_coef[c][1][2], t1, z);
            z = fmaf(s_coef[c][2][0], h2, z);
            z = fmaf(s_coef[c][2][1], r2, z);
            z = fmaf(s_coef[c][2][2], t2, z);
            z = fmaxf(z, 0.0f);
            acc = fmaf(s_fc[c * 62 + i], z, acc);
        }
        h0 = h1; h1 = h2;
        r0 = r1; r1 = r2;
        t0 = t1; t1 = t2;
    }

    const float ev = (acc > 0.0f) ? acc : 0.01f * acc;   // leaky_relu(0.01)
    eArr[e] = ev;
    atomicMaxFloat(&m[head], ev);
}

// ---------------------------------------------------------------------------
// K3: ex = exp(e - m[head]); denom[head] += ex (atomic)
// ---------------------------------------------------------------------------
__global__ void edge_exp_kernel(const int* __restrict__ triple,
                                float* __restrict__ eArr,
                                const float* __restrict__ m,
                                float* __restrict__ denom, int E) {
    const int e = blockIdx.x * blockDim.x + threadIdx.x;
    if (e >= E) return;
    const int head = triple[3 * e + 0];
    const float ex = expf(eArr[e] - m[head]);
    eArr[e] = ex;               // overwrite in place (same thread, same slot)
    atomicAdd(&denom[head], ex);
}

// ---------------------------------------------------------------------------
// K4: agg[head] += (ex/denom[head]) * inp[tail]  (64 f32 atomics per edge)
// ---------------------------------------------------------------------------
__global__ void edge_agg_kernel(const int* __restrict__ triple,
                                const float* __restrict__ eArr,
                                const float* __restrict__ denom,
                                const float* __restrict__ inp,
                                float* __restrict__ agg, int E) {
    const int e = blockIdx.x * blockDim.x + threadIdx.x;
    if (e >= E) return;
    const int head = triple[3 * e + 0];
    const int tail = triple[3 * e + 2];
    const float attn = eArr[e] / denom[head];
    const float* tp = inp + (size_t)tail * OUT_DIM;
    float* ap = agg + (size_t)head * OUT_DIM;
#pragma unroll
    for (int j = 0; j < OUT_DIM; ++j)
        atomicAdd(&ap[j], attn * tp[j]);
}

// ---------------------------------------------------------------------------
// K5: out = elu(agg + inp)
// ---------------------------------------------------------------------------
__global__ void finish_kernel(const float* __restrict__ inp,
                              const float* __restrict__ agg,
                              float* __restrict__ out, int total) {
    const int idx = blockIdx.x * blockDim.x + threadIdx.x;
    if (idx >= total) return;
    const float v = agg[idx] + inp[idx];
    out[idx] = (v > 0.0f) ? v : (expf(v) - 1.0f);
}

// ---------------------------------------------------------------------------
extern "C" void kernel_launch(void* const* d_in, const int* in_sizes, int n_in,
                              void* d_out, int out_size, void* d_ws, size_t ws_size,
                              hipStream_t stream) {
    const float* input     = (const float*)d_in[0];
    const int*   triple    = (const int*)d_in[1];
    const float* W         = (const float*)d_in[2];
    const float* rel_embed = (const float*)d_in[3];
    const float* conv_w    = (const float*)d_in[4];
    const float* conv_b    = (const float*)d_in[5];
    const float* fc_w      = (const float*)d_in[6];
    const float* bn1_gamma = (const float*)d_in[7];
    const float* bn1_beta  = (const float*)d_in[8];
    const float* bn2_gamma = (const float*)d_in[9];
    const float* bn2_beta  = (const float*)d_in[10];

    const int N = in_sizes[0] / IN_DIM;
    const int E = in_sizes[1] / 3;

    // workspace layout (floats): inp | e/ex | m | denom | agg
    float* ws    = (float*)d_ws;
    float* inp   = ws;                         // N * 64
    float* eArr  = inp + (size_t)N * OUT_DIM;  // E
    float* m     = eArr + E;                   // N
    float* denom = m + N;                      // N
    float* agg   = denom + N;                  // N * 64

    const int total = N * OUT_DIM;

    init_ws_kernel<<<(total + 255) / 256, 256, 0, stream>>>(m, denom, agg, N);

    // GEMM: 8 waves/block, 16 rows per wave
    const int rowTiles = (N + 15) / 16;
    gemm_wmma_kernel<<<(rowTiles + 7) / 8, 256, 0, stream>>>(input, W, inp, N);

    const int eBlocks = (E + 255) / 256;
    edge_score_kernel<<<eBlocks, 256, 0, stream>>>(
        triple, inp, rel_embed, conv_w, conv_b, fc_w,
        bn1_gamma, bn1_beta, bn2_gamma, bn2_beta, eArr, m, E);

    edge_exp_kernel<<<eBlocks, 256, 0, stream>>>(triple, eArr, m, denom, E);

    edge_agg_kernel<<<eBlocks, 256, 0, stream>>>(triple, eArr, denom, inp, agg, E);

    finish_kernel<<<(total + 255) / 256, 256, 0, stream>>>(inp, agg, (float*)d_out, total);
}